// Autoencoder_72894184948286
// MI455X (gfx1250) — compile-verified
//
#include <hip/hip_runtime.h>
#include <stdint.h>

// ---------------------------------------------------------------------------
// 2-layer GCN (PyG GCNConv style) + FC head, for MI455X (gfx1250, wave32).
//   deg -> rsqrt -> [GEMM(WMMA f32) -> bias-init -> edge gather/scatter ->
//   ReLU] x2 -> FC.
// GEMMs use V_WMMA_F32_16X16X4_F32 (exact fp32; compute is not the
// bottleneck).  Aggregation is wave-per-edge with float4 gathers and
// global f32 atomics (L2-resident working set ~103MB < 192MB L2).
// ---------------------------------------------------------------------------

typedef float v2f __attribute__((ext_vector_type(2)));
typedef float v8f __attribute__((ext_vector_type(8)));

#define DIN 128
#define DH1 256   // conv1 output width (2*hidden)
#define DH2 128   // conv2 output width
#define DOUT 6

// ---------------- degree ----------------
__global__ void k_init_deg(float* deg, int n) {
  int i = blockIdx.x * blockDim.x + threadIdx.x;
  if (i < n) deg[i] = 1.0f;              // self-loop contributes 1
}

__global__ void k_count_deg(const long long* __restrict__ ei, float* deg, int E) {
  int e = blockIdx.x * blockDim.x + threadIdx.x;
  if (e < E) atomicAdd(&deg[(int)ei[(size_t)E + e]], 1.0f);
}

__global__ void k_deg_isqrt(float* deg, int n) {
  int i = blockIdx.x * blockDim.x + threadIdx.x;
  if (i < n) deg[i] = rsqrtf(deg[i]);    // deg >= 1 always (self-loop)
}

// ---------------- elementwise ----------------
// agg[i] = bias[i % D]  (D is a power of two)
__global__ void k_init_bias(float* __restrict__ agg, const float* __restrict__ b,
                            int total, int D) {
  for (int i = blockIdx.x * blockDim.x + threadIdx.x; i < total;
       i += gridDim.x * blockDim.x)
    agg[i] = b[i & (D - 1)];
}

__global__ void k_relu(float* __restrict__ v, int total) {
  for (int i = blockIdx.x * blockDim.x + threadIdx.x; i < total;
       i += gridDim.x * blockDim.x)
    v[i] = fmaxf(v[i], 0.0f);
}

// ---------------- fp32 WMMA GEMM: C[M x NC] = A[M x K] @ B[K x NC] ----------
// One wave per 16x16 C tile, 8 waves/block share an A row-strip.
// Requires M % 16 == 0 and NC % (16*8 per blockIdx.y granule) handled by grid.
__global__ __launch_bounds__(256) void k_gemm_wmma(
    const float* __restrict__ A, const float* __restrict__ B,
    float* __restrict__ C, int K, int NC) {
  const int wave  = threadIdx.x >> 5;
  const int lane  = threadIdx.x & 31;
  const int half  = lane >> 4;     // 0: lanes 0-15, 1: lanes 16-31
  const int lrow  = lane & 15;
  const int tileM = blockIdx.x;
  const int tileN = blockIdx.y * 8 + wave;

  const int row = tileM * 16 + lrow;         // A row for this lane
  const int col = tileN * 16 + lrow;         // B/C column for this lane

  v8f c = {0.f, 0.f, 0.f, 0.f, 0.f, 0.f, 0.f, 0.f};

  const float* Arow = A + (size_t)row * K;
#pragma unroll 4
  for (int k0 = 0; k0 < K; k0 += 4) {
    const int ka = k0 + 2 * half;            // A 16x4 f32 layout: low lanes
    v2f a, b;                                // hold K=0,1; high lanes K=2,3
    a.x = Arow[ka];
    a.y = Arow[ka + 1];
    b.x = B[(size_t)ka * NC + col];          // B 4x16 mirrored layout
    b.y = B[(size_t)(ka + 1) * NC + col];
    c = __builtin_amdgcn_wmma_f32_16x16x4_f32(
        /*neg_a=*/false, a, /*neg_b=*/false, b,
        /*c_mod=*/(short)0, c, /*reuse_a=*/false, /*reuse_b=*/false);
  }

  // C/D 16x16 f32 layout: VGPR r -> (M=r, N=lane) | (M=r+8, N=lane-16)
  float* Cblk = C + (size_t)(tileM * 16) * NC + tileN * 16;
#pragma unroll
  for (int r = 0; r < 8; ++r) {
    const int m = r + 8 * half;
    Cblk[(size_t)m * NC + lrow] = c[r];
  }
}

// ---------------- edge gather/scatter (wave per edge) ----------------
// agg[dst] += norm(src,dst) * h[src]  for all E edges plus N self-loops.
template <int D>
__global__ __launch_bounds__(256) void k_message(
    const float* __restrict__ h, const long long* __restrict__ ei,
    const float* __restrict__ dis, float* __restrict__ agg, int E, int n) {
  const int gid  = blockIdx.x * blockDim.x + threadIdx.x;
  const int edge = gid >> 5;
  const int lane = gid & 31;
  const int total = E + n;
  if (edge >= total) return;

  long long s, d;
  if (edge < E) {
    s = ei[edge];
    d = ei[(size_t)E + edge];
  } else {
    s = d = (long long)(edge - E);           // self-loop
  }
  const float norm = dis[s] * dis[d];

  // each lane moves D/32 contiguous floats (D/128 float4 chunks)
  const float4* hs = (const float4*)(h + (size_t)s * D) + lane * (D / 128);
  float* ad = agg + (size_t)d * D + lane * (D / 32);
#pragma unroll
  for (int c = 0; c < D / 128; ++c) {
    const float4 v = hs[c];
    atomicAdd(ad + c * 4 + 0, v.x * norm);
    atomicAdd(ad + c * 4 + 1, v.y * norm);
    atomicAdd(ad + c * 4 + 2, v.z * norm);
    atomicAdd(ad + c * 4 + 3, v.w * norm);
  }
}

// ---------------- final FC: out[N x 6] = h2 @ Wfc + bfc ----------------
__global__ void k_fc(const float* __restrict__ h, const float* __restrict__ W,
                     const float* __restrict__ b, float* __restrict__ out, int n) {
  int i = blockIdx.x * blockDim.x + threadIdx.x;
  if (i >= n) return;
  float acc[DOUT];
#pragma unroll
  for (int j = 0; j < DOUT; ++j) acc[j] = b[j];
  const float* hr = h + (size_t)i * DH2;
  for (int k = 0; k < DH2; ++k) {
    const float a = hr[k];
#pragma unroll
    for (int j = 0; j < DOUT; ++j) acc[j] = fmaf(a, W[k * DOUT + j], acc[j]);
  }
#pragma unroll
  for (int j = 0; j < DOUT; ++j) out[(size_t)i * DOUT + j] = acc[j];
}

// ---------------------------------------------------------------------------
extern "C" void kernel_launch(void* const* d_in, const int* in_sizes, int n_in,
                              void* d_out, int out_size, void* d_ws, size_t ws_size,
                              hipStream_t stream) {
  const float*     x   = (const float*)d_in[0];
  const float*     W1  = (const float*)d_in[1];
  const float*     b1  = (const float*)d_in[2];
  const float*     W2  = (const float*)d_in[3];
  const float*     b2  = (const float*)d_in[4];
  const float*     Wfc = (const float*)d_in[5];
  const float*     bfc = (const float*)d_in[6];
  const long long* ei  = (const long long*)d_in[7];   // int64 edge_index [2, E]

  const int n = in_sizes[0] / DIN;   // 50000 (multiple of 16)
  const int E = in_sizes[7] / 2;     // 500000

  // Workspace layout (floats), with aliasing to keep footprint ~103 MB:
  //   [0, n)                        dis  (degree -> deg^-1/2)
  //   [n, n + n*256)                h1   ; later reused as h2   (n*128)
  //   [n + n*256, n + 2*n*256)      agg1 ; later reused as agg2 (n*128)
  float* ws   = (float*)d_ws;
  float* dis  = ws;
  float* h1   = ws + n;
  float* agg1 = ws + n + (size_t)n * DH1;
  float* h2   = h1;    // h1 dead after msg1
  float* agg2 = agg1;  // agg1 dead after gemm2 (init_agg2 runs after gemm2)

  const int tB = 256;

  // ---- symmetric-normalized degrees ----
  k_init_deg<<<(n + tB - 1) / tB, tB, 0, stream>>>(dis, n);
  k_count_deg<<<(E + tB - 1) / tB, tB, 0, stream>>>(ei, dis, E);
  k_deg_isqrt<<<(n + tB - 1) / tB, tB, 0, stream>>>(dis, n);

  // ---- layer 1: h1 = x @ W1 ; agg1 = b1 + sum_edges norm*h1[src] ; ReLU ----
  k_gemm_wmma<<<dim3(n / 16, DH1 / 128), tB, 0, stream>>>(x, W1, h1, DIN, DH1);
  {
    const int tot = n * DH1;
    k_init_bias<<<(tot + tB - 1) / tB, tB, 0, stream>>>(agg1, b1, tot, DH1);
    const int thr = (E + n) * 32;  // one wave32 per edge
    k_message<DH1><<<(thr + tB - 1) / tB, tB, 0, stream>>>(h1, ei, dis, agg1, E, n);
    k_relu<<<(tot + tB - 1) / tB, tB, 0, stream>>>(agg1, tot);
  }

  // ---- layer 2: h2 = agg1 @ W2 ; agg2 = b2 + aggregate ; ReLU ----
  k_gemm_wmma<<<dim3(n / 16, DH2 / 128), tB, 0, stream>>>(agg1, W2, h2, DH1, DH2);
  {
    const int tot = n * DH2;
    k_init_bias<<<(tot + tB - 1) / tB, tB, 0, stream>>>(agg2, b2, tot, DH2);
    const int thr = (E + n) * 32;
    k_message<DH2><<<(thr + tB - 1) / tB, tB, 0, stream>>>(h2, ei, dis, agg2, E, n);
    k_relu<<<(tot + tB - 1) / tB, tB, 0, stream>>>(agg2, tot);
  }

  // ---- FC head ----
  k_fc<<<(n + tB - 1) / tB, tB, 0, stream>>>(agg2, Wfc, bfc, (float*)d_out, n);
}